// CombinedLoss_49907519979675
// MI455X (gfx1250) — compile-verified
//
#include <hip/hip_runtime.h>
#include <hip/hip_bf16.h>

typedef __attribute__((ext_vector_type(2))) float v2f;
typedef __attribute__((ext_vector_type(8))) float v8f;

#define NB   1024        // batch
#define ND   768         // feature dim
#define NC   100000      // classes
#define EPS  0.1f
#define MARG 0.3f

// ---------------------------------------------------------------------------
// Workspace layout (floats / uints), 16 KB total:
//   ws[0    ..1023] : per-row CE contribution
//   ws[1024 ..2047] : dist_ap bits (atomicMax, u32)
//   ws[2048 ..3071] : dist_an bits (atomicMin, u32)
//   ws[3072 ..4095] : sq[i] = ||x_i||^2
// ---------------------------------------------------------------------------

__global__ void init_kernel(unsigned* ap_bits, unsigned* an_bits) {
    int i = blockIdx.x * blockDim.x + threadIdx.x;
    if (i < NB) {
        ap_bits[i] = 0u;           // +0.0f (diag dist is always a valid positive)
        an_bits[i] = 0x7F800000u;  // +inf
    }
}

// One block per row: sq[i] = sum_k x[i][k]^2
__global__ void sqnorm_kernel(const float* __restrict__ x, float* __restrict__ sq) {
    __shared__ float red[256];
    const int row = blockIdx.x;
    const float* p = x + (size_t)row * ND;
    float s = 0.f;
    for (int k = threadIdx.x; k < ND; k += 256) {
        float v = p[k];
        s += v * v;
    }
    red[threadIdx.x] = s;
    __syncthreads();
    for (int off = 128; off > 0; off >>= 1) {
        if (threadIdx.x < off) red[threadIdx.x] += red[threadIdx.x + off];
        __syncthreads();
    }
    if (threadIdx.x == 0) sq[row] = red[0];
}

// ---------------------------------------------------------------------------
// Gram matrix via V_WMMA_F32_16X16X4_F32 + fused batch-hard mining.
// Block = 256 threads = 8 waves arranged 2x4; block tile = 32 rows x 64 cols.
// K staged through LDS in chunks of 32 (padded stride 33 vs 64 banks).
// ---------------------------------------------------------------------------
#define KC 32

__global__ void __launch_bounds__(256)
gram_mine_kernel(const float* __restrict__ x, const int* __restrict__ labels,
                 const float* __restrict__ sq,
                 unsigned* __restrict__ ap_bits, unsigned* __restrict__ an_bits) {
    __shared__ float Alds[32][KC + 1];
    __shared__ float Blds[64][KC + 1];

    const int tid  = threadIdx.x;
    const int lane = tid & 31;
    const int wid  = tid >> 5;
    const int wr   = wid >> 2;    // 0..1  tile-row within block
    const int wc   = wid & 3;     // 0..3  tile-col within block
    const int half = lane >> 4;   // 0..1
    const int mn   = lane & 15;   // row/col index within 16

    const int row0 = blockIdx.y * 32;   // A rows covered by this block
    const int col0 = blockIdx.x * 64;   // B rows (= output cols) covered

    v8f acc = {};

    for (int k0 = 0; k0 < ND; k0 += KC) {
        // cooperative coalesced stage: A 32x32, B 64x32
        #pragma unroll
        for (int it = 0; it < 4; ++it) {
            int idx = tid + it * 256;
            int r = idx >> 5, k = idx & 31;
            Alds[r][k] = x[(size_t)(row0 + r) * ND + k0 + k];
        }
        #pragma unroll
        for (int it = 0; it < 8; ++it) {
            int idx = tid + it * 256;
            int r = idx >> 5, k = idx & 31;
            Blds[r][k] = x[(size_t)(col0 + r) * ND + k0 + k];
        }
        if (k0 + KC < ND)  // hint next chunk into cache -> global_prefetch_b8
            __builtin_prefetch(&x[(size_t)(row0 + (tid >> 3)) * ND + k0 + KC + (tid & 7) * 4], 0, 1);
        __syncthreads();

        const int am = wr * 16 + mn;   // A-frag row for this lane
        const int bn = wc * 16 + mn;   // B-frag col (stored as row of X) for this lane
        #pragma unroll
        for (int kk = 0; kk < KC; kk += 4) {
            v2f a, b;
            // A 16x4 f32 layout: VGPR0 -> K = 2*half, VGPR1 -> K = 2*half+1
            a.x = Alds[am][kk + 2 * half];
            a.y = Alds[am][kk + 2 * half + 1];
            // B 4x16 f32 layout mirrored: B[k][n] = X[col0+n][k]
            b.x = Blds[bn][kk + 2 * half];
            b.y = Blds[bn][kk + 2 * half + 1];
            acc = __builtin_amdgcn_wmma_f32_16x16x4_f32(
                false, a, false, b, (short)0, acc, false, false);
        }
        __syncthreads();
    }

    // ---- fused mining epilogue ----
    const int jcol = col0 + wc * 16 + mn;          // this lane's output column
    const int labj = labels[jcol];
    const float sqj = sq[jcol];

    #pragma unroll
    for (int r = 0; r < 8; ++r) {
        const int irow = row0 + wr * 16 + 8 * half + r;   // D VGPR r -> M = r + 8*half
        const float g   = acc[r];
        const float d2  = sq[irow] + sqj - 2.0f * g;
        const float dist = sqrtf(fmaxf(d2, 1e-12f));
        const bool  pos  = (labels[irow] == labj);

        float apv = pos ? dist : 0.0f;
        float anv = pos ? __builtin_inff() : dist;
        #pragma unroll
        for (int off = 1; off < 16; off <<= 1) {   // reduce across the 16-lane half
            apv = fmaxf(apv, __shfl_xor(apv, off, 32));
            anv = fminf(anv, __shfl_xor(anv, off, 32));
        }
        if (mn == 0) {   // one lane per (half, r) commits this row's stripe result
            atomicMax(&ap_bits[irow], __float_as_uint(apv));
            atomicMin(&an_bits[irow], __float_as_uint(anv));
        }
    }
}

// ---------------------------------------------------------------------------
// Label-smoothed CE: one block per row, streaming online logsumexp + sum(x).
// r_b = lse - (1-eps)*x[label] - (eps/C)*sum_x ; id_loss = mean_b r_b
// ---------------------------------------------------------------------------
__global__ void __launch_bounds__(256)
ce_row_kernel(const float* __restrict__ cls, const int* __restrict__ labels,
              float* __restrict__ rb) {
    __shared__ float sm[256], ss[256], sx[256];
    const int row = blockIdx.x;
    const float4* p = (const float4*)(cls + (size_t)row * NC);
    const int nv = NC / 4;   // 25000

    float m = -__builtin_inff(), s = 0.f, sumx = 0.f;
    for (int i = threadIdx.x; i < nv; i += 256) {
        float4 v = p[i];
        float e[4] = {v.x, v.y, v.z, v.w};
        #pragma unroll
        for (int q = 0; q < 4; ++q) {
            float xv = e[q];
            sumx += xv;
            if (xv > m) { s = s * __expf(m - xv) + 1.0f; m = xv; }
            else        { s += __expf(xv - m); }
        }
    }
    sm[threadIdx.x] = m; ss[threadIdx.x] = s; sx[threadIdx.x] = sumx;
    __syncthreads();
    for (int off = 128; off > 0; off >>= 1) {
        if (threadIdx.x < off) {
            float m2 = sm[threadIdx.x + off], s2 = ss[threadIdx.x + off];
            float mn2 = fmaxf(sm[threadIdx.x], m2);
            ss[threadIdx.x] = ss[threadIdx.x] * __expf(sm[threadIdx.x] - mn2)
                            + s2 * __expf(m2 - mn2);
            sm[threadIdx.x] = mn2;
            sx[threadIdx.x] += sx[threadIdx.x + off];
        }
        __syncthreads();
    }
    if (threadIdx.x == 0) {
        float lse = sm[0] + __logf(ss[0]);
        float xl  = cls[(size_t)row * NC + labels[row]];
        rb[row] = lse - (1.0f - EPS) * xl - (EPS / (float)NC) * sx[0];
    }
}

// ---------------------------------------------------------------------------
// Final reduce: id_loss = mean(rb); triplet = mean(relu(ap - an + margin))
// d_out = { loss, id_loss, triplet }
// ---------------------------------------------------------------------------
__global__ void finalize_kernel(const float* __restrict__ rb,
                                const unsigned* __restrict__ ap_bits,
                                const unsigned* __restrict__ an_bits,
                                float* __restrict__ out) {
    __shared__ float r1[256], r2[256];
    float a = 0.f, t = 0.f;
    for (int i = threadIdx.x; i < NB; i += 256) {
        a += rb[i];
        float d = __uint_as_float(ap_bits[i]) - __uint_as_float(an_bits[i]) + MARG;
        t += fmaxf(d, 0.0f);   // relu(-inf) = 0 handles no-negative rows
    }
    r1[threadIdx.x] = a; r2[threadIdx.x] = t;
    __syncthreads();
    for (int off = 128; off > 0; off >>= 1) {
        if (threadIdx.x < off) {
            r1[threadIdx.x] += r1[threadIdx.x + off];
            r2[threadIdx.x] += r2[threadIdx.x + off];
        }
        __syncthreads();
    }
    if (threadIdx.x == 0) {
        float id_loss = r1[0] / (float)NB;
        float tri     = r2[0] / (float)NB;
        out[0] = id_loss + tri;
        out[1] = id_loss;
        out[2] = tri;
    }
}

extern "C" void kernel_launch(void* const* d_in, const int* in_sizes, int n_in,
                              void* d_out, int out_size, void* d_ws, size_t ws_size,
                              hipStream_t stream) {
    const float* cls    = (const float*)d_in[0];   // [1024, 100000] fp32
    const float* gfeat  = (const float*)d_in[1];   // [1024, 768]   fp32
    // d_in[2] (feat) unused by the forward pass
    const int*   labels = (const int*)d_in[3];     // [1024] int32
    float* out = (float*)d_out;                    // {loss, id_loss, triplet}

    float*    rb      = (float*)d_ws;
    unsigned* ap_bits = (unsigned*)d_ws + 1024;
    unsigned* an_bits = (unsigned*)d_ws + 2048;
    float*    sq      = (float*)d_ws + 3072;

    init_kernel<<<dim3(NB / 256), dim3(256), 0, stream>>>(ap_bits, an_bits);
    sqnorm_kernel<<<dim3(NB), dim3(256), 0, stream>>>(gfeat, sq);
    gram_mine_kernel<<<dim3(NB / 64, NB / 32), dim3(256), 0, stream>>>(
        gfeat, labels, sq, ap_bits, an_bits);
    ce_row_kernel<<<dim3(NB), dim3(256), 0, stream>>>(cls, labels, rb);
    finalize_kernel<<<dim3(1), dim3(256), 0, stream>>>(rb, ap_bits, an_bits, out);
}